// GATLayerTemporal_14516989460844
// MI455X (gfx1250) — compile-verified
//
#include <hip/hip_runtime.h>

// GAT temporal layer, MI455X (gfx1250, wave32).
// Fixed shapes: N=4, H=W=16, T=24, V=32.
// Roofline: ~6 MB total traffic, ~150 MFLOP -> launch/latency bound; keep fp32
// and use the native fp32 matrix op V_WMMA_F32_16X16X4_F32 for the V-dim
// contractions (exact precision, no conversion traffic).

#define ALPHA 0.2f
#define N_  4
#define HD  16
#define WD  16
#define T_  24
#define V_  32
#define HW  256     // HD*WD
#define THW 512     // 2*HD*WD

typedef __attribute__((ext_vector_type(2))) float v2f;
typedef __attribute__((ext_vector_type(8))) float v8f;

// ---------------------------------------------------------------------------
// Kernel A: adj = B + 1e-6 + I, min-max normalized, D^-1/2 A D^-1/2.
// One wave, thread i owns row i.
// ---------------------------------------------------------------------------
__global__ void adj_kernel(const float* __restrict__ B, float* __restrict__ adjn) {
    __shared__ float rmin[V_], rmax[V_], dinv[V_];
    __shared__ float smn, smx;
    int i = threadIdx.x;
    float mn = 1e30f, mx = -1e30f;
    for (int j = 0; j < V_; ++j) {
        float v = B[i * V_ + j] + 1e-6f + (i == j ? 1.0f : 0.0f);
        mn = fminf(mn, v); mx = fmaxf(mx, v);
    }
    rmin[i] = mn; rmax[i] = mx;
    __syncthreads();
    if (i == 0) {
        float gmn = rmin[0], gmx = rmax[0];
        for (int k = 1; k < V_; ++k) { gmn = fminf(gmn, rmin[k]); gmx = fmaxf(gmx, rmax[k]); }
        smn = gmn; smx = gmx;
    }
    __syncthreads();
    float gmn = smn, inv = 1.0f / (smx - smn);
    float s = 0.0f;
    for (int j = 0; j < V_; ++j) {
        float v = B[i * V_ + j] + 1e-6f + (i == j ? 1.0f : 0.0f);
        s += (v - gmn) * inv;
    }
    dinv[i] = rsqrtf(s);
    __syncthreads();
    float di = dinv[i];
    for (int j = 0; j < V_; ++j) {
        float v = B[i * V_ + j] + 1e-6f + (i == j ? 1.0f : 0.0f);
        adjn[i * V_ + j] = (v - gmn) * inv * di * dinv[j];
    }
}

// ---------------------------------------------------------------------------
// Kernel B: Wh[n,t,v,p] = sum_t' h[n,p,t',v] * W[t',t].
// One block per (n,p); 24x32 tile staged in LDS, broadcast-friendly reads.
// Wh stored [N][T][V][HW] so per-(n,t) GEMM slices are contiguous.
// ---------------------------------------------------------------------------
__global__ void __launch_bounds__(256) wh_kernel(const float* __restrict__ h,
                                                 const float* __restrict__ Wp,
                                                 float* __restrict__ Wh) {
    __shared__ float Xs[T_][V_];   // [t'][v]
    __shared__ float Wm[T_][T_];   // [t'][t]
    int bid = blockIdx.x;
    int n = bid >> 8;
    int p = bid & 255;
    int tid = threadIdx.x;
    const float* hp = h + (size_t)(n * HW + p) * T_ * V_;
    for (int idx = tid; idx < T_ * V_; idx += 256)
        Xs[idx / V_][idx & (V_ - 1)] = hp[idx];
    for (int idx = tid; idx < T_ * T_; idx += 256)
        Wm[idx / T_][idx % T_] = Wp[idx];
    __syncthreads();
    for (int idx = tid; idx < V_ * T_; idx += 256) {
        int v = idx / T_, t = idx % T_;
        float acc = 0.0f;
#pragma unroll
        for (int tp = 0; tp < T_; ++tp) acc += Xs[tp][v] * Wm[tp][t];
        Wh[(size_t)((n * T_ + t) * V_ + v) * HW + p] = acc;
    }
}

// ---------------------------------------------------------------------------
// Kernel C: the torch-view attention projection, decomposed.
// flat f = h*768 + w*24 + t over (H,2W,T); row t_new = f/512, p = f%512.
// w<16 contributes to ei (depends only on i), w>=16 to ej (only on j).
// ei/ej: [N][V][T], one thread per (n,v,tn) element.
// ---------------------------------------------------------------------------
__global__ void eij_kernel(const float* __restrict__ Wh, const float* __restrict__ a,
                           float* __restrict__ ei, float* __restrict__ ej) {
    int g = blockIdx.x * 256 + threadIdx.x;     // N*V*T = 3072 exactly
    int n  = g / (V_ * T_);
    int r  = g - n * (V_ * T_);
    int vv = r / T_;
    int tn = r - vv * T_;
    const float* whb = Wh + (size_t)n * T_ * V_ * HW;
    float ai = 0.0f, aj = 0.0f;
    for (int q = 0; q < THW; ++q) {
        int f  = tn * THW + q;
        int hh = f / (2 * WD * T_);             // /768
        int r2 = f - hh * (2 * WD * T_);
        int ww = r2 / T_;
        int t  = r2 - ww * T_;
        float w = a[q];
        if (ww < WD)
            ai += whb[(size_t)(t * V_ + vv) * HW + hh * WD + ww] * w;
        else
            aj += whb[(size_t)(t * V_ + vv) * HW + hh * WD + (ww - WD)] * w;
    }
    ei[g] = ai;     // layout [n][v][t] == flat g
    ej[g] = aj;
}

// ---------------------------------------------------------------------------
// Kernel D: e = leakyrelu(ei+ej), softmax over T. att stored [N][T][V][V]
// so each (n,t) slice is a contiguous 32x32 matrix for the WMMA stage.
// ---------------------------------------------------------------------------
__global__ void att_kernel(const float* __restrict__ ei, const float* __restrict__ ej,
                           float* __restrict__ att) {
    int g = blockIdx.x * 256 + threadIdx.x;     // N*V*V = 4096 exactly
    int n   = g >> 10;
    int rem = g & 1023;
    int i   = rem >> 5;
    int j   = rem & 31;
    const float* eip = ei + (n * V_ + i) * T_;
    const float* ejp = ej + (n * V_ + j) * T_;
    float l[T_];
    float m = -1e30f;
#pragma unroll
    for (int t = 0; t < T_; ++t) {
        float e = eip[t] + ejp[t];
        e = e > 0.0f ? e : ALPHA * e;
        l[t] = e;
        m = fmaxf(m, e);
    }
    float s = 0.0f;
#pragma unroll
    for (int t = 0; t < T_; ++t) { l[t] = __expf(l[t] - m); s += l[t]; }
    float is = 1.0f / s;
#pragma unroll
    for (int t = 0; t < T_; ++t)
        att[(size_t)((n * T_ + t) * V_ + i) * V_ + j] = l[t] * is;
}

// ---------------------------------------------------------------------------
// Kernel E: fused h2 = (adjn^T @ att @ Wh), then sigmoid(elu(.)).
// One block per (n,t): M = adjn^T * G (32x32, VALU in LDS), then
// O(32x256) = M @ Wslice via V_WMMA_F32_16X16X4_F32:
//   2x16 output tiles, K=32 -> 8 wmma k-steps each; 8 waves x 4 tiles.
// A frag (16x4 f32, ISA layout): lanes 0-15 K=0,1 ; lanes 16-31 K=2,3.
// B frag (4x16 f32): VGPR0 -> K = lanehalf*2, VGPR1 -> K = lanehalf*2+1.
// C/D frag: VGPR r -> row tm*16 + r + lanehalf*8, col tn*16 + lane%16.
// EXEC stays all-ones: no divergence around the WMMA.
// ---------------------------------------------------------------------------
__global__ void __launch_bounds__(256) gemm_kernel(const float* __restrict__ Wh,
                                                   const float* __restrict__ att,
                                                   const float* __restrict__ adjn,
                                                   float* __restrict__ out) {
    __shared__ float Gl [V_ * 33];     // att slice, padded
    __shared__ float Ajl[V_ * 33];     // adjn, padded
    __shared__ float Ml [V_ * 33];     // M = adjn^T * G
    __shared__ float Wl [V_ * 257];    // Wh slice 32x256, padded row stride

    int bid = blockIdx.x;              // N*T = 96
    int n = bid / T_;
    int t = bid - n * T_;
    int tid = threadIdx.x;
    const float* gsl = att + (size_t)(n * T_ + t) * V_ * V_;
    const float* wsl = Wh  + (size_t)(n * T_ + t) * V_ * HW;

    for (int idx = tid; idx < V_ * V_; idx += 256) {
        Gl [(idx >> 5) * 33 + (idx & 31)] = gsl[idx];
        Ajl[(idx >> 5) * 33 + (idx & 31)] = adjn[idx];
    }
    for (int idx = tid; idx < V_ * HW; idx += 256)
        Wl[(idx >> 8) * 257 + (idx & 255)] = wsl[idx];
    __syncthreads();

    // M[v][j] = sum_i adjn[i][v] * G[i][j]
    for (int idx = tid; idx < V_ * V_; idx += 256) {
        int v = idx >> 5, j = idx & 31;
        float acc = 0.0f;
#pragma unroll
        for (int i2 = 0; i2 < V_; ++i2) acc += Ajl[i2 * 33 + v] * Gl[i2 * 33 + j];
        Ml[v * 33 + j] = acc;
    }
    __syncthreads();

    int wave = tid >> 5, lane = tid & 31;
    int lh = lane >> 4, lm = lane & 15;
    for (int q = 0; q < 4; ++q) {
        int tile = wave * 4 + q;       // 32 tiles, uniform over waves
        int tm = tile >> 4, tn = tile & 15;
        int arow = tm * 16 + lm;
        int col  = tn * 16 + lm;
        v8f c = {};
#pragma unroll
        for (int kk = 0; kk < 8; ++kk) {
            int kb = kk * 4 + lh * 2;
            v2f a, b;
            a.x = Ml[arow * 33 + kb];
            a.y = Ml[arow * 33 + kb + 1];
            b.x = Wl[kb * 257 + col];
            b.y = Wl[(kb + 1) * 257 + col];
            c = __builtin_amdgcn_wmma_f32_16x16x4_f32(
                    /*neg_a=*/false, a, /*neg_b=*/false, b,
                    /*c_mod=*/(short)0, c, /*reuse_a=*/false, /*reuse_b=*/false);
        }
        // out[n][p][t][v] = sigmoid(elu(O[v][p]))
        int vb = tm * 16 + lh * 8;
        float* ob = out + ((size_t)(n * HW + col) * T_ + t) * V_ + vb;
#pragma unroll
        for (int r = 0; r < 8; ++r) {
            float x = c[r];
            float y = x > 0.0f ? x : (__expf(x) - 1.0f);   // elu, alpha=1
            ob[r] = 1.0f / (1.0f + __expf(-y));            // sigmoid
        }
    }
}

// ---------------------------------------------------------------------------
extern "C" void kernel_launch(void* const* d_in, const int* in_sizes, int n_in,
                              void* d_out, int out_size, void* d_ws, size_t ws_size,
                              hipStream_t stream) {
    const float* h  = (const float*)d_in[0];   // (N,H,W,T,V)
    const float* Wp = (const float*)d_in[1];   // (T,T)
    const float* ap = (const float*)d_in[2];   // (512,1)
    const float* Bp = (const float*)d_in[3];   // (V,V)
    float* out = (float*)d_out;

    float* ws    = (float*)d_ws;
    float* Whb   = ws;                               // N*T*V*HW = 786432
    float* adjn  = Whb  + (size_t)N_ * T_ * V_ * HW; // 1024
    float* ei    = adjn + V_ * V_;                   // 3072
    float* ej    = ei   + N_ * V_ * T_;              // 3072
    float* attb  = ej   + N_ * V_ * T_;              // 98304  (~3.5 MB total)

    adj_kernel <<<1,           32, 0, stream>>>(Bp, adjn);
    wh_kernel  <<<N_ * HW,    256, 0, stream>>>(h, Wp, Whb);
    eij_kernel <<<(N_ * V_ * T_) / 256, 256, 0, stream>>>(Whb, ap, ei, ej);
    att_kernel <<<(N_ * V_ * V_) / 256, 256, 0, stream>>>(ei, ej, attb);
    gemm_kernel<<<N_ * T_,    256, 0, stream>>>(Whb, attb, adjn, out);
}